// GCN_37658273251498
// MI455X (gfx1250) — compile-verified
//
#include <hip/hip_runtime.h>
#include <math.h>

typedef float v2f __attribute__((ext_vector_type(2)));
typedef float v8f __attribute__((ext_vector_type(8)));

// ---------------- utility kernels ----------------

__global__ void __launch_bounds__(256) k_fill(float* __restrict__ p, float v, int n) {
  int i = blockIdx.x * 256 + threadIdx.x;
  if (i < n) p[i] = v;
}

__global__ void __launch_bounds__(256) k_deg_count(const int* __restrict__ dst,
                                                   float* __restrict__ deg, int e) {
  int i = blockIdx.x * 256 + threadIdx.x;
  if (i < e) unsafeAtomicAdd(&deg[dst[i]], 1.0f);
}

__global__ void __launch_bounds__(256) k_rsqrt_inplace(float* __restrict__ p, int n) {
  int i = blockIdx.x * 256 + threadIdx.x;
  if (i < n) p[i] = rsqrtf(p[i]);  // deg >= 1 always (self loop counted)
}

// ---------------- dense GEMM: C[N,128] = A[N,128] @ W[128,128] ----------------
// One wave (32 lanes) per 16x16 output tile, V_WMMA_F32_16X16X4_F32, K-loop of 32.
// Block = 256 threads = 8 waves covering a 16-row x 128-col strip.
// A frag (16x4 f32): lane m=lane&15 is row M; lane>>4 selects K pair {0,1}/{2,3}.
// B frag (4x16 f32): lane&15 is col N; lane>>4 selects K rows {0,1}/{2,3}.
// C/D frag: VGPR j -> row (j + 8*(lane>>4)), col lane&15.
__global__ void __launch_bounds__(256) k_gemm128_wmma(const float* __restrict__ A,
                                                      const float* __restrict__ W,
                                                      float* __restrict__ C) {
  const int lane = threadIdx.x & 31;
  const int wave = threadIdx.x >> 5;
  const int row0 = blockIdx.x << 4;
  const int col0 = wave << 4;
  const int m    = lane & 15;
  const int kh   = (lane >> 4) << 1;  // 0 or 2
  const float* arow = A + (size_t)(row0 + m) * 128;
  v8f acc = {};
#pragma unroll
  for (int k = 0; k < 128; k += 4) {
    v2f a, b;
    a[0] = arow[k + kh];
    a[1] = arow[k + kh + 1];
    b[0] = W[(size_t)(k + kh) * 128 + col0 + m];
    b[1] = W[(size_t)(k + kh + 1) * 128 + col0 + m];
    acc = __builtin_amdgcn_wmma_f32_16x16x4_f32(false, a, false, b, (short)0, acc,
                                                false, false);
  }
  const int rbase = row0 + ((lane >> 4) << 3);
#pragma unroll
  for (int j = 0; j < 8; ++j)
    C[(size_t)(rbase + j) * 128 + col0 + m] = acc[j];
}

// ---------------- edge scatter (128-wide): agg[dst] += h[src] * norm ----------------
// One wave per edge; lane handles 4 contiguous floats (16B load + 4 f32 atomics).
__global__ void __launch_bounds__(256) k_scatter128(const float* __restrict__ h,
                                                    const int* __restrict__ src,
                                                    const int* __restrict__ dst,
                                                    const float* __restrict__ dinv,
                                                    float* __restrict__ agg, int e) {
  int widx = blockIdx.x * 8 + (threadIdx.x >> 5);
  if (widx >= e) return;
  int lane = threadIdx.x & 31;
  int s = src[widx];
  int d = dst[widx];
  float nrm = dinv[s] * dinv[d];
  float4 hv = ((const float4*)(h + (size_t)s * 128))[lane];
  float* o = agg + (size_t)d * 128 + lane * 4;
  unsafeAtomicAdd(o + 0, hv.x * nrm);
  unsafeAtomicAdd(o + 1, hv.y * nrm);
  unsafeAtomicAdd(o + 2, hv.z * nrm);
  unsafeAtomicAdd(o + 3, hv.w * nrm);
}

// ---------------- edge scatter, narrow feature dim (8 or 1) ----------------
__global__ void __launch_bounds__(256) k_scatter_small(const float* __restrict__ h,
                                                       const int* __restrict__ src,
                                                       const int* __restrict__ dst,
                                                       const float* __restrict__ dinv,
                                                       float* __restrict__ agg,
                                                       int e, int cdim) {
  int t = blockIdx.x * 256 + threadIdx.x;
  if (t >= e * cdim) return;
  int ei = t / cdim;
  int c  = t - ei * cdim;
  int s = src[ei];
  int d = dst[ei];
  float nrm = dinv[s] * dinv[d];
  unsafeAtomicAdd(&agg[(size_t)d * cdim + c], h[(size_t)s * cdim + c] * nrm);
}

// ---------------- narrow GEMM: H[N,cdim] = A[N,128] @ W[128,cdim] ----------------
__global__ void __launch_bounds__(256) k_gemm_small(const float* __restrict__ A,
                                                    const float* __restrict__ W,
                                                    float* __restrict__ H, int n, int cdim) {
  int t = blockIdx.x * 256 + threadIdx.x;
  if (t >= n * cdim) return;
  int row = t / cdim;
  int c   = t - row * cdim;
  const float* a = A + (size_t)row * 128;
  float s = 0.0f;
#pragma unroll 8
  for (int k = 0; k < 128; ++k) s = fmaf(a[k], W[(size_t)k * cdim + c], s);
  H[t] = s;
}

// ---------------- finalize: out = [relu](agg + h*dinv^2 + bias) ----------------
// self-loop term h[i]*dinv[i]^2 folded in here.
__global__ void __launch_bounds__(256) k_finalize(const float* __restrict__ agg,
                                                  const float* __restrict__ h,
                                                  const float* __restrict__ dinv,
                                                  const float* __restrict__ bias,
                                                  float* __restrict__ out,
                                                  int n, int cdim, int relu) {
  int t = blockIdx.x * 256 + threadIdx.x;
  if (t >= n * cdim) return;
  int row = t / cdim;
  int c   = t - row * cdim;
  float di = dinv[row];
  float v  = agg[t] + h[t] * di * di + bias[c];
  out[t] = relu ? fmaxf(v, 0.0f) : v;
}

// ---------------- host side ----------------

static inline int cdiv(long a, long b) { return (int)((a + b - 1) / b); }

extern "C" void kernel_launch(void* const* d_in, const int* in_sizes, int n_in,
                              void* d_out, int out_size, void* d_ws, size_t ws_size,
                              hipStream_t stream) {
  const float* x   = (const float*)d_in[0];
  const int*   ei  = (const int*)d_in[1];
  const float* Wp1 = (const float*)d_in[2];  const float* bp1 = (const float*)d_in[3];
  const float* Wp2 = (const float*)d_in[4];  const float* bp2 = (const float*)d_in[5];
  const float* Wp3 = (const float*)d_in[6];  const float* bp3 = (const float*)d_in[7];
  const float* Wv1 = (const float*)d_in[8];  const float* bv1 = (const float*)d_in[9];
  const float* Wv2 = (const float*)d_in[10]; const float* bv2 = (const float*)d_in[11];
  const float* Wv3 = (const float*)d_in[12]; const float* bv3 = (const float*)d_in[13];

  const int n = in_sizes[0] / 128;   // 50000 (divisible by 16)
  const int e = in_sizes[1] / 2;     // 600000
  const int* src = ei;               // edge_index[0]
  const int* dst = ei + e;           // edge_index[1]

  // workspace layout (f32): dinv[n] | bufH[n*128] | bufA[n*128]  (~51.5 MB)
  float* dinv = (float*)d_ws;
  float* bufH = dinv + n;
  float* bufA = bufH + (size_t)n * 128;

  float* logits = (float*)d_out;               // [n, 8]
  float* value  = logits + (size_t)n * 8;      // [n, 1]

  const int nf = n * 128;
  const int gemmBlocks = n / 16;
  const int scatBlocks = cdiv(e, 8);

  // ---- degrees -> dinv (deg starts at 1 for the self loop) ----
  k_fill<<<cdiv(n, 256), 256, 0, stream>>>(dinv, 1.0f, n);
  k_deg_count<<<cdiv(e, 256), 256, 0, stream>>>(dst, dinv, e);
  k_rsqrt_inplace<<<cdiv(n, 256), 256, 0, stream>>>(dinv, n);

  // one 128->128 GCN conv; stream order makes in-place bufA reuse safe:
  // gemm(in) completes before bufA (== in on later layers) is zeroed.
  auto conv128 = [&](const float* in, const float* W, const float* b, int relu) {
    k_gemm128_wmma<<<gemmBlocks, 256, 0, stream>>>(in, W, bufH);
    k_fill<<<cdiv(nf, 256), 256, 0, stream>>>(bufA, 0.0f, nf);
    k_scatter128<<<scatBlocks, 256, 0, stream>>>(bufH, src, dst, dinv, bufA, e);
    k_finalize<<<cdiv(nf, 256), 256, 0, stream>>>(bufA, bufH, dinv, b, bufA, n, 128, relu);
  };
  auto conv_small = [&](const float* in, const float* W, const float* b, int cdim,
                        float* out) {
    k_gemm_small<<<cdiv((long)n * cdim, 256), 256, 0, stream>>>(in, W, bufH, n, cdim);
    k_fill<<<cdiv((long)n * cdim, 256), 256, 0, stream>>>(bufA, 0.0f, n * cdim);
    k_scatter_small<<<cdiv((long)e * cdim, 256), 256, 0, stream>>>(bufH, src, dst, dinv,
                                                                   bufA, e, cdim);
    k_finalize<<<cdiv((long)n * cdim, 256), 256, 0, stream>>>(bufA, bufH, dinv, b, out,
                                                              n, cdim, 0);
  };

  // ---- policy branch ----
  conv128(x,    Wp1, bp1, 1);
  conv128(bufA, Wp2, bp2, 1);
  conv_small(bufA, Wp3, bp3, 8, logits);

  // ---- value branch ----
  conv128(x,    Wv1, bv1, 1);
  conv128(bufA, Wv2, bv2, 1);
  conv_small(bufA, Wv3, bv3, 1, value);
}